// VoxtralAttention_3393024164130
// MI455X (gfx1250) — compile-verified
//
#include <hip/hip_runtime.h>

typedef __attribute__((ext_vector_type(2))) float v2f;
typedef __attribute__((ext_vector_type(8))) float v8f;

#define T_TOK 2048
#define HID_DIM 4096
#define NH 32
#define NKV 8
#define HD 128
#define QKV_COLS ((NH + 2 * NKV) * HD)   // 6144
#define LOG_BASE 13.815510557964274f     // ln(1e6)

// ---- gfx1250 async global->LDS copies (guarded; fall back to reg staging) ----
#if defined(__has_builtin)
#if __has_builtin(__builtin_amdgcn_global_load_async_to_lds_b128) && \
    __has_builtin(__builtin_amdgcn_s_wait_asynccnt)
#define HAVE_ASYNC 1
#endif
#endif
#ifndef HAVE_ASYNC
#define HAVE_ASYNC 0
#endif

#if HAVE_ASYNC
typedef int v4i_async __attribute__((vector_size(16)));
typedef __attribute__((address_space(1))) v4i_async* gptr_b128;
typedef __attribute__((address_space(3))) v4i_async* lptr_b128;
#define ASYNC_B128(gsrc, ldst)                                            \
  __builtin_amdgcn_global_load_async_to_lds_b128(                         \
      (gptr_b128)(gsrc), (lptr_b128)(ldst), 0, 0)
#define WAIT_ASYNC() __builtin_amdgcn_s_wait_asynccnt(0)
#endif

// ---------------------------------------------------------------------------
// Generic fp32 WMMA GEMM: C[M,N] = A[M,K] * B[K,N]
// Block tile 128x128, BK=16, 256 threads = 8 waves (2x4), wave tile 64x32.
// Double-buffered LDS with async global->LDS copies when available.
// ---------------------------------------------------------------------------
#define AS_STRIDE 24   // 16 + pad, keeps 16B alignment
#define BS_STRIDE 132  // 128 + pad

__global__ __launch_bounds__(256)
void gemm_f32_wmma(const float* __restrict__ A, const float* __restrict__ B,
                   float* __restrict__ C, int M, int N, int K) {
  __shared__ float As[2][128 * AS_STRIDE];
  __shared__ float Bs[2][16 * BS_STRIDE];

  const int tid  = threadIdx.x;
  const int lane = tid & 31;
  const int wave = tid >> 5;      // 0..7
  const int wm   = wave >> 2;     // 0..1
  const int wn   = wave & 3;      // 0..3
  const int half = lane >> 4;     // selects K pair (ISA fp32 A/B layout)
  const int l16  = lane & 15;

  const int bm = blockIdx.y * 128;
  const int bn = blockIdx.x * 128;

  v8f acc[4][2];
#pragma unroll
  for (int mt = 0; mt < 4; ++mt)
#pragma unroll
    for (int nt = 0; nt < 2; ++nt)
      acc[mt][nt] = (v8f){0.f, 0.f, 0.f, 0.f, 0.f, 0.f, 0.f, 0.f};

  const int arow = tid >> 1;            // 0..127
  const int acol = (tid & 1) * 8;       // 0 or 8
  const int brow = tid >> 4;            // 0..15
  const int bcol = (tid & 15) * 8;      // 0..120

  auto compute_tile = [&](const float* as, const float* bs) {
#pragma unroll
    for (int kk = 0; kk < 16; kk += 4) {
      const int ka = kk + half * 2;
      v2f af[4], bf[2];
#pragma unroll
      for (int mt = 0; mt < 4; ++mt)
        af[mt] = *(const v2f*)&as[(wm * 64 + mt * 16 + l16) * AS_STRIDE + ka];
#pragma unroll
      for (int nt = 0; nt < 2; ++nt) {
        const int cn = wn * 32 + nt * 16 + l16;
        v2f b;
        b.x = bs[ka * BS_STRIDE + cn];
        b.y = bs[(ka + 1) * BS_STRIDE + cn];
        bf[nt] = b;
      }
#pragma unroll
      for (int mt = 0; mt < 4; ++mt)
#pragma unroll
        for (int nt = 0; nt < 2; ++nt)
          acc[mt][nt] = __builtin_amdgcn_wmma_f32_16x16x4_f32(
              false, af[mt], false, bf[nt], (short)0, acc[mt][nt], false, false);
    }
  };

#if HAVE_ASYNC
  auto issue_tile = [&](int k0, int buf) {
    const float* ag = A + (size_t)(bm + arow) * K + k0 + acol;
    ASYNC_B128(ag,     &As[buf][arow * AS_STRIDE + acol]);
    ASYNC_B128(ag + 4, &As[buf][arow * AS_STRIDE + acol + 4]);
    const float* bg = B + (size_t)(k0 + brow) * N + bn + bcol;
    ASYNC_B128(bg,     &Bs[buf][brow * BS_STRIDE + bcol]);
    ASYNC_B128(bg + 4, &Bs[buf][brow * BS_STRIDE + bcol + 4]);
  };

  issue_tile(0, 0);
  WAIT_ASYNC();
  __syncthreads();
  for (int k0 = 0; k0 < K; k0 += 16) {
    const int cur = (k0 >> 4) & 1;
    if (k0 + 16 < K) issue_tile(k0 + 16, cur ^ 1);
    compute_tile(&As[cur][0], &Bs[cur][0]);
    WAIT_ASYNC();
    __syncthreads();
  }
#else
  for (int k0 = 0; k0 < K; k0 += 16) {
    {
      const float* ag = A + (size_t)(bm + arow) * K + k0 + acol;
      float4 a0 = *(const float4*)(ag);
      float4 a1 = *(const float4*)(ag + 4);
      *(float4*)&As[0][arow * AS_STRIDE + acol]     = a0;
      *(float4*)&As[0][arow * AS_STRIDE + acol + 4] = a1;
      if (k0 + 16 < K) __builtin_prefetch(ag + 16, 0, 0);
    }
    {
      const float* bg = B + (size_t)(k0 + brow) * N + bn + bcol;
      float4 b0 = *(const float4*)(bg);
      float4 b1 = *(const float4*)(bg + 4);
      *(float4*)&Bs[0][brow * BS_STRIDE + bcol]     = b0;
      *(float4*)&Bs[0][brow * BS_STRIDE + bcol + 4] = b1;
      if (k0 + 16 < K) __builtin_prefetch(bg + (size_t)16 * N, 0, 0);
    }
    __syncthreads();
    compute_tile(&As[0][0], &Bs[0][0]);
    __syncthreads();
  }
#endif

  // Store: C layout — VGPR r, lanes 0-15 -> M=r, lanes 16-31 -> M=r+8
#pragma unroll
  for (int mt = 0; mt < 4; ++mt) {
#pragma unroll
    for (int nt = 0; nt < 2; ++nt) {
      const int row0 = bm + wm * 64 + mt * 16 + half * 8;
      const int col  = bn + wn * 32 + nt * 16 + l16;
#pragma unroll
      for (int r = 0; r < 8; ++r)
        C[(size_t)(row0 + r) * N + col] = acc[mt][nt][r];
    }
  }
}

// ---------------------------------------------------------------------------
// GPT-J interleaved RoPE on q and k (contiguous columns [0, (NH+NKV)*HD))
// ---------------------------------------------------------------------------
__global__ __launch_bounds__(256)
void rope_kernel(const int* __restrict__ positions, float* __restrict__ qkv) {
  const int idx = blockIdx.x * blockDim.x + threadIdx.x;
  const int d2   = idx & 63;
  const int rest = idx >> 6;
  const int hh   = rest % (NH + NKV);
  const int t    = rest / (NH + NKV);
  if (t >= T_TOK) return;

  const float pos = (float)positions[t];
  const float inv_freq = __expf(-LOG_BASE * ((float)(2 * d2) / (float)HD));
  const float f = pos * inv_freq;
  const float c = __cosf(f);
  const float s = __sinf(f);

  const size_t base = (size_t)t * QKV_COLS + hh * HD;
  const float x1 = qkv[base + 2 * d2];
  const float x2 = qkv[base + 2 * d2 + 1];
  qkv[base + 2 * d2]     = x1 * c - x2 * s;
  qkv[base + 2 * d2 + 1] = x2 * c + x1 * s;
}

// ---------------------------------------------------------------------------
// Flash attention: block = (64 q rows, 1 head), 4 waves x 16 rows.
// Online softmax, causal, GQA. Double-buffered K/V via async LDS copies.
// ---------------------------------------------------------------------------
#define KS_STRIDE 132
#define PS_STRIDE 20

__global__ __launch_bounds__(128)
void attn_kernel(const float* __restrict__ qkv, float* __restrict__ attn) {
  __shared__ float Ks[2][16 * KS_STRIDE];
  __shared__ float Vs[2][16 * KS_STRIDE];
  __shared__ float Ps[4 * 16 * PS_STRIDE];

  const int tid  = threadIdx.x;
  const int lane = tid & 31;
  const int wave = tid >> 5;
  const int half = lane >> 4;
  const int l16  = lane & 15;

  const int h   = blockIdx.y;
  const int kvh = h >> 2;
  const int q0  = blockIdx.x * 64;
  const int qw  = q0 + wave * 16;

  // Q tile 16x128 as 32 fp32 A-fragments
  v2f qf[32];
  {
    const float* qp = qkv + (size_t)(qw + l16) * QKV_COLS + h * HD;
#pragma unroll
    for (int kk = 0; kk < 32; ++kk)
      qf[kk] = *(const v2f*)(qp + kk * 4 + half * 2);
  }

  v8f o[8];
#pragma unroll
  for (int nt = 0; nt < 8; ++nt)
    o[nt] = (v8f){0.f, 0.f, 0.f, 0.f, 0.f, 0.f, 0.f, 0.f};
  float mr[8], lr[8];
#pragma unroll
  for (int r = 0; r < 8; ++r) { mr[r] = -3.402823466e38f; lr[r] = 0.f; }

  const float scale = 0.08838834764831845f;   // 1/sqrt(128)
  const int kb_max = (q0 + 63) >> 4;

  const int ldrow = tid >> 3;           // 0..15
  const int ldcol = (tid & 7) * 16;     // 0..112

#if HAVE_ASYNC
  auto issue_kv = [&](int kb, int buf) {
    const float* kg = qkv + (size_t)(kb * 16 + ldrow) * QKV_COLS + NH * HD + kvh * HD + ldcol;
    const float* vg = qkv + (size_t)(kb * 16 + ldrow) * QKV_COLS + (NH + NKV) * HD + kvh * HD + ldcol;
#pragma unroll
    for (int i = 0; i < 4; ++i) {
      ASYNC_B128(kg + i * 4, &Ks[buf][ldrow * KS_STRIDE + ldcol + i * 4]);
      ASYNC_B128(vg + i * 4, &Vs[buf][ldrow * KS_STRIDE + ldcol + i * 4]);
    }
  };
  issue_kv(0, 0);
  WAIT_ASYNC();
  __syncthreads();
#endif

  for (int kb = 0; kb <= kb_max; ++kb) {
#if HAVE_ASYNC
    const int cur = kb & 1;
    if (kb + 1 <= kb_max) issue_kv(kb + 1, cur ^ 1);
    const float* ks = &Ks[cur][0];
    const float* vs = &Vs[cur][0];
#else
    {
      const float* kg = qkv + (size_t)(kb * 16 + ldrow) * QKV_COLS + NH * HD + kvh * HD + ldcol;
      const float* vg = qkv + (size_t)(kb * 16 + ldrow) * QKV_COLS + (NH + NKV) * HD + kvh * HD + ldcol;
#pragma unroll
      for (int i = 0; i < 4; ++i) {
        *(float4*)&Ks[0][ldrow * KS_STRIDE + ldcol + i * 4] = *(const float4*)(kg + i * 4);
        *(float4*)&Vs[0][ldrow * KS_STRIDE + ldcol + i * 4] = *(const float4*)(vg + i * 4);
      }
    }
    __syncthreads();
    const float* ks = &Ks[0][0];
    const float* vs = &Vs[0][0];
#endif

    // S = Q * K^T
    v8f s = (v8f){0.f, 0.f, 0.f, 0.f, 0.f, 0.f, 0.f, 0.f};
#pragma unroll
    for (int kk = 0; kk < 32; ++kk) {
      v2f bf = *(const v2f*)&ks[l16 * KS_STRIDE + kk * 4 + half * 2];
      s = __builtin_amdgcn_wmma_f32_16x16x4_f32(
          false, qf[kk], false, bf, (short)0, s, false, false);
    }

    // Online softmax (each row lives in one 16-lane half)
#pragma unroll
    for (int r = 0; r < 8; ++r) {
      const int gi = qw + r + half * 8;
      const int gj = kb * 16 + l16;
      float v = s[r] * scale;
      if (gj > gi) v = -3.402823466e38f;
      float rm = v;
      rm = fmaxf(rm, __shfl_xor(rm, 1));
      rm = fmaxf(rm, __shfl_xor(rm, 2));
      rm = fmaxf(rm, __shfl_xor(rm, 4));
      rm = fmaxf(rm, __shfl_xor(rm, 8));
      const float mnew  = fmaxf(mr[r], rm);
      const float alpha = __expf(mr[r] - mnew);
      const float p     = __expf(v - mnew);
      float rs = p;
      rs += __shfl_xor(rs, 1);
      rs += __shfl_xor(rs, 2);
      rs += __shfl_xor(rs, 4);
      rs += __shfl_xor(rs, 8);
      lr[r] = lr[r] * alpha + rs;
      mr[r] = mnew;
#pragma unroll
      for (int nt = 0; nt < 8; ++nt) o[nt][r] *= alpha;
      Ps[(wave * 16 + r + half * 8) * PS_STRIDE + l16] = p;   // C -> LDS
    }

    // Reload P as A-fragments (wave-private; DS in-order per wave)
    v2f pf[4];
#pragma unroll
    for (int kk = 0; kk < 4; ++kk)
      pf[kk] = *(const v2f*)&Ps[(wave * 16 + l16) * PS_STRIDE + kk * 4 + half * 2];

    // O += P * V
#pragma unroll
    for (int nt = 0; nt < 8; ++nt) {
#pragma unroll
      for (int kk = 0; kk < 4; ++kk) {
        const int krow = kk * 4 + half * 2;
        v2f bf;
        bf.x = vs[krow * KS_STRIDE + nt * 16 + l16];
        bf.y = vs[(krow + 1) * KS_STRIDE + nt * 16 + l16];
        o[nt] = __builtin_amdgcn_wmma_f32_16x16x4_f32(
            false, pf[kk], false, bf, (short)0, o[nt], false, false);
      }
    }
#if HAVE_ASYNC
    WAIT_ASYNC();
#endif
    __syncthreads();
  }

  // Epilogue
#pragma unroll
  for (int r = 0; r < 8; ++r) {
    const float inv = 1.0f / lr[r];
    const int row = qw + r + half * 8;
#pragma unroll
    for (int nt = 0; nt < 8; ++nt)
      attn[(size_t)row * (NH * HD) + h * HD + nt * 16 + l16] = o[nt][r] * inv;
  }
}

// ---------------------------------------------------------------------------
extern "C" void kernel_launch(void* const* d_in, const int* in_sizes, int n_in,
                              void* d_out, int out_size, void* d_ws, size_t ws_size,
                              hipStream_t stream) {
  const int*   positions = (const int*)d_in[0];
  const float* hidden    = (const float*)d_in[1];
  const float* Wqkv      = (const float*)d_in[2];
  const float* Wo        = (const float*)d_in[3];
  float* out = (float*)d_out;

  float* qkv  = (float*)d_ws;                          // T x 6144  (48 MB)
  float* attn = qkv + (size_t)T_TOK * QKV_COLS;        // T x 4096  (32 MB)

  {
    dim3 grid(QKV_COLS / 128, T_TOK / 128);
    gemm_f32_wmma<<<grid, 256, 0, stream>>>(hidden, Wqkv, qkv, T_TOK, QKV_COLS, HID_DIM);
  }
  {
    const int total = T_TOK * (NH + NKV) * (HD / 2);
    rope_kernel<<<total / 256, 256, 0, stream>>>(positions, qkv);
  }
  {
    dim3 grid(T_TOK / 64, NH);
    attn_kernel<<<grid, 128, 0, stream>>>(qkv, attn);
  }
  {
    dim3 grid(HID_DIM / 128, T_TOK / 128);
    gemm_f32_wmma<<<grid, 256, 0, stream>>>(attn, Wo, out, T_TOK, HID_DIM, HID_DIM);
  }
}